// MinGRU_53893249630393
// MI455X (gfx1250) — compile-verified
//
#include <hip/hip_runtime.h>
#include <math.h>

// ---------------------------------------------------------------------------
// MinGRU forward for MI455X (gfx1250, wave32, WMMA).
//   hidden = x @ Wh ; gate = x @ Wg          (fused fp32 WMMA GEMM)
//   log-space Heinsen scan along S           (online logcumsumexp)
//   out = h @ Wo                             (fp32 WMMA GEMM)
// Shapes: B=4, S=4096, D=1024, Di=1536, M = B*S = 16384.
//
// GEMM design: 256 threads = 8 waves, workgroup tile 64(M) x 128(N),
// K-tile 16, register double-buffered LDS staging. B tiles stored in LDS
// K-pair interleaved so every WMMA B fragment is ONE aligned ds_load_b64
// (no v_dual_mov shuffling). A tile row-major padded (stride 20 floats).
// ---------------------------------------------------------------------------

typedef __attribute__((ext_vector_type(2))) float v2f;
typedef __attribute__((ext_vector_type(8))) float v8f;

#define TM 64     // workgroup M tile
#define TN 128    // workgroup N tile
#define TK 16     // K tile
#define LDA 20    // A LDS row stride (floats): 80 B, 16B aligned
#define LDP 264   // B LDS k-pair row stride (floats): 2*TN + 8, 1056 B

__device__ __forceinline__ float softplus_f(float x) {
  // stable log(1 + exp(x))
  return fmaxf(x, 0.0f) + log1pf(__expf(-fabsf(x)));
}

__device__ __forceinline__ v8f wmma_f32(v2f a, v2f b, v8f c) {
  // V_WMMA_F32_16X16X4_F32 : D = A(16x4) * B(4x16) + C(16x16), fp32
  return __builtin_amdgcn_wmma_f32_16x16x4_f32(
      /*neg_a=*/false, a, /*neg_b=*/false, b,
      /*c_mod=*/(short)0, c, /*reuse_a=*/false, /*reuse_b=*/false);
}

// ---------------------------------------------------------------------------
// Kernel 1: fused hidden/gate projection + elementwise log-space transforms.
//   X [M x K], Wh/Wg [K x N] row-major.
//   lcb = log_coeffs = -softplus(gate)
//   lvb = log_values = -softplus(-gate) + log_g(hidden)
// ---------------------------------------------------------------------------
__global__ __launch_bounds__(256) void mingru_gemm_hg(
    const float* __restrict__ X, const float* __restrict__ Wh,
    const float* __restrict__ Wg, float* __restrict__ lcb,
    float* __restrict__ lvb, int M, int K, int N) {
  __shared__ __align__(16) float sA[2][TM * LDA];          // 2 * 64*20*4 B
  __shared__ __align__(16) float sBh[2][(TK / 2) * LDP];   // 2 * 8*264*4 B
  __shared__ __align__(16) float sBg[2][(TK / 2) * LDP];

  const int ntiles = N / TN;
  const int m0 = (blockIdx.x / ntiles) * TM;
  const int n0 = (blockIdx.x % ntiles) * TN;
  const int tid  = threadIdx.x;
  const int lane = tid & 31;
  const int wv   = tid >> 5;   // 8 waves
  const int wm   = wv & 1;     // 2 wave rows (32 M each)
  const int wn   = wv >> 1;    // 4 wave cols (32 N each)
  const int half = lane >> 4;
  const int l15  = lane & 15;
  const int c0   = wn * 32 + l15;
  const int c1   = c0 + 16;

  // staging task mapping (exactly 1 task per thread per tile)
  const int ar  = tid >> 2;        // A row 0..63
  const int ac4 = tid & 3;         // A float4 col 0..3
  const int bkp = tid >> 5;        // B k-pair 0..7
  const int bc4 = tid & 31;        // B float4 col 0..31

  const v8f vzero = {0.f, 0.f, 0.f, 0.f, 0.f, 0.f, 0.f, 0.f};
  v8f accH[2][2], accG[2][2];
#pragma unroll
  for (int i = 0; i < 2; ++i)
#pragma unroll
    for (int j = 0; j < 2; ++j) { accH[i][j] = vzero; accG[i][j] = vzero; }

  float4 ra, rh0, rh1, rg0, rg1;   // staging registers (tile k+1)

  auto load_tile = [&](int k0) {
    ra  = *(const float4*)&X[(size_t)(m0 + ar) * K + k0 + ac4 * 4];
    const size_t gb = (size_t)(k0 + 2 * bkp) * N + n0 + bc4 * 4;
    rh0 = *(const float4*)&Wh[gb];
    rh1 = *(const float4*)&Wh[gb + N];
    rg0 = *(const float4*)&Wg[gb];
    rg1 = *(const float4*)&Wg[gb + N];
  };
  auto store_tile = [&](int buf) {
    *(float4*)&sA[buf][ar * LDA + ac4 * 4] = ra;
    float* bh = &sBh[buf][bkp * LDP + bc4 * 8];
    float* bg = &sBg[buf][bkp * LDP + bc4 * 8];
    v2f t;
    t.x = rh0.x; t.y = rh1.x; ((v2f*)bh)[0] = t;
    t.x = rh0.y; t.y = rh1.y; ((v2f*)bh)[1] = t;
    t.x = rh0.z; t.y = rh1.z; ((v2f*)bh)[2] = t;
    t.x = rh0.w; t.y = rh1.w; ((v2f*)bh)[3] = t;
    t.x = rg0.x; t.y = rg1.x; ((v2f*)bg)[0] = t;
    t.x = rg0.y; t.y = rg1.y; ((v2f*)bg)[1] = t;
    t.x = rg0.z; t.y = rg1.z; ((v2f*)bg)[2] = t;
    t.x = rg0.w; t.y = rg1.w; ((v2f*)bg)[3] = t;
  };

  const int nk = K / TK;
  load_tile(0);
  for (int kt = 0; kt < nk; ++kt) {
    const int buf = kt & 1;
    store_tile(buf);
    __syncthreads();
    if (kt + 1 < nk) load_tile((kt + 1) * TK);     // overlap with compute
    if (kt + 4 < nk) {                              // gfx1250 global_prefetch_b8
      const int kf = (kt + 4) * TK;
      __builtin_prefetch(&X[(size_t)(m0 + ar) * K + kf + ac4 * 4], 0, 3);
      __builtin_prefetch(&Wh[(size_t)(kf + 2 * bkp) * N + n0 + bc4 * 4], 0, 3);
      __builtin_prefetch(&Wg[(size_t)(kf + 2 * bkp) * N + n0 + bc4 * 4], 0, 3);
    }
#pragma unroll
    for (int kk = 0; kk < TK; kk += 4) {
      const int kp = (kk >> 1) + half;              // even k-row parity ensured
      v2f a0  = *(const v2f*)&sA[buf][(wm * 32 + l15) * LDA + kk + 2 * half];
      v2f a1  = *(const v2f*)&sA[buf][(wm * 32 + 16 + l15) * LDA + kk + 2 * half];
      v2f bh0 = *(const v2f*)&sBh[buf][kp * LDP + c0 * 2];
      v2f bh1 = *(const v2f*)&sBh[buf][kp * LDP + c1 * 2];
      v2f bg0 = *(const v2f*)&sBg[buf][kp * LDP + c0 * 2];
      v2f bg1 = *(const v2f*)&sBg[buf][kp * LDP + c1 * 2];

      accH[0][0] = wmma_f32(a0, bh0, accH[0][0]);
      accH[0][1] = wmma_f32(a0, bh1, accH[0][1]);
      accH[1][0] = wmma_f32(a1, bh0, accH[1][0]);
      accH[1][1] = wmma_f32(a1, bh1, accH[1][1]);
      accG[0][0] = wmma_f32(a0, bg0, accG[0][0]);
      accG[0][1] = wmma_f32(a0, bg1, accG[0][1]);
      accG[1][0] = wmma_f32(a1, bg0, accG[1][0]);
      accG[1][1] = wmma_f32(a1, bg1, accG[1][1]);
    }
  }

  // --- epilogue: elementwise log-space transforms, write lc / lv ---
#pragma unroll
  for (int sm = 0; sm < 2; ++sm) {
#pragma unroll
    for (int sn = 0; sn < 2; ++sn) {
      const int col = n0 + wn * 32 + sn * 16 + l15;
#pragma unroll
      for (int r = 0; r < 8; ++r) {
        const int row = m0 + wm * 32 + sm * 16 + half * 8 + r;
        float Hv = accH[sm][sn][r];
        float Gv = accG[sm][sn][r];
        float lc = -softplus_f(Gv);                 // log(1 - z)
        float lz = -softplus_f(-Gv);                // log z
        float lth = (Hv >= 0.0f) ? __logf(Hv + 0.5f) : -softplus_f(-Hv);
        const size_t o = (size_t)row * N + col;
        lcb[o] = lc;
        lvb[o] = lz + lth;
      }
    }
  }
}

// ---------------------------------------------------------------------------
// Kernel 2: log-space scan along S. One lane per (b, channel) chain.
// h_t = exp(A_t + Mx_t + log(Se_t)), written in place over the lc buffer.
// Load addresses are data-independent -> they pipeline ahead of the VALU
// dependence chain; speculative prefetch runs 16 steps ahead.
// ---------------------------------------------------------------------------
__global__ __launch_bounds__(256) void mingru_scan(
    float* __restrict__ lcb, const float* __restrict__ lvb,
    int Bq, int S, int Di) {
  const int c = blockIdx.x * blockDim.x + threadIdx.x;
  if (c >= Bq * Di) return;
  const int b = c / Di;
  const int f = c - b * Di;
  const size_t base = (size_t)b * S * Di + f;

  float A = 0.0f;            // cumsum of log_coeffs (a_star)
  float Mx = -INFINITY;      // running max of (lv - A)
  float Se = 0.0f;           // running sum of exp(x - Mx)
  for (int t = 0; t < S; ++t) {
    const size_t idx = base + (size_t)t * Di;
    __builtin_prefetch(&lcb[idx + (size_t)16 * Di], 0, 3);
    __builtin_prefetch(&lvb[idx + (size_t)16 * Di], 0, 3);
    const float lc = lcb[idx];
    const float lv = lvb[idx];
    A += lc;
    const float x = lv - A;
    if (x > Mx) {
      Se = Se * __expf(Mx - x) + 1.0f;  // first iter: 0*exp(-inf)+1 = 1
      Mx = x;
    } else {
      Se += __expf(x - Mx);
    }
    lcb[idx] = __expf(A + Mx + __logf(Se));  // h_t
  }
}

// ---------------------------------------------------------------------------
// Kernel 3: out = h @ Wo.  h [M x K], Wo [K x N] row-major.
// ---------------------------------------------------------------------------
__global__ __launch_bounds__(256) void mingru_gemm_out(
    const float* __restrict__ H, const float* __restrict__ Wo,
    float* __restrict__ out, int M, int K, int N) {
  __shared__ __align__(16) float sA[2][TM * LDA];
  __shared__ __align__(16) float sB[2][(TK / 2) * LDP];

  const int ntiles = N / TN;
  const int m0 = (blockIdx.x / ntiles) * TM;
  const int n0 = (blockIdx.x % ntiles) * TN;
  const int tid  = threadIdx.x;
  const int lane = tid & 31;
  const int wv   = tid >> 5;
  const int wm   = wv & 1;
  const int wn   = wv >> 1;
  const int half = lane >> 4;
  const int l15  = lane & 15;
  const int c0   = wn * 32 + l15;
  const int c1   = c0 + 16;

  const int ar  = tid >> 2;
  const int ac4 = tid & 3;
  const int bkp = tid >> 5;
  const int bc4 = tid & 31;

  const v8f vzero = {0.f, 0.f, 0.f, 0.f, 0.f, 0.f, 0.f, 0.f};
  v8f acc[2][2];
#pragma unroll
  for (int i = 0; i < 2; ++i)
#pragma unroll
    for (int j = 0; j < 2; ++j) acc[i][j] = vzero;

  float4 ra, rb0, rb1;

  auto load_tile = [&](int k0) {
    ra  = *(const float4*)&H[(size_t)(m0 + ar) * K + k0 + ac4 * 4];
    const size_t gb = (size_t)(k0 + 2 * bkp) * N + n0 + bc4 * 4;
    rb0 = *(const float4*)&Wo[gb];
    rb1 = *(const float4*)&Wo[gb + N];
  };
  auto store_tile = [&](int buf) {
    *(float4*)&sA[buf][ar * LDA + ac4 * 4] = ra;
    float* bp = &sB[buf][bkp * LDP + bc4 * 8];
    v2f t;
    t.x = rb0.x; t.y = rb1.x; ((v2f*)bp)[0] = t;
    t.x = rb0.y; t.y = rb1.y; ((v2f*)bp)[1] = t;
    t.x = rb0.z; t.y = rb1.z; ((v2f*)bp)[2] = t;
    t.x = rb0.w; t.y = rb1.w; ((v2f*)bp)[3] = t;
  };

  const int nk = K / TK;
  load_tile(0);
  for (int kt = 0; kt < nk; ++kt) {
    const int buf = kt & 1;
    store_tile(buf);
    __syncthreads();
    if (kt + 1 < nk) load_tile((kt + 1) * TK);
    if (kt + 4 < nk) {
      const int kf = (kt + 4) * TK;
      __builtin_prefetch(&H[(size_t)(m0 + ar) * K + kf + ac4 * 4], 0, 3);
      __builtin_prefetch(&Wo[(size_t)(kf + 2 * bkp) * N + n0 + bc4 * 4], 0, 3);
    }
#pragma unroll
    for (int kk = 0; kk < TK; kk += 4) {
      const int kp = (kk >> 1) + half;
      v2f a0 = *(const v2f*)&sA[buf][(wm * 32 + l15) * LDA + kk + 2 * half];
      v2f a1 = *(const v2f*)&sA[buf][(wm * 32 + 16 + l15) * LDA + kk + 2 * half];
      v2f b0 = *(const v2f*)&sB[buf][kp * LDP + c0 * 2];
      v2f b1 = *(const v2f*)&sB[buf][kp * LDP + c1 * 2];

      acc[0][0] = wmma_f32(a0, b0, acc[0][0]);
      acc[0][1] = wmma_f32(a0, b1, acc[0][1]);
      acc[1][0] = wmma_f32(a1, b0, acc[1][0]);
      acc[1][1] = wmma_f32(a1, b1, acc[1][1]);
    }
  }

#pragma unroll
  for (int sm = 0; sm < 2; ++sm) {
#pragma unroll
    for (int sn = 0; sn < 2; ++sn) {
      const int col = n0 + wn * 32 + sn * 16 + l15;
#pragma unroll
      for (int r = 0; r < 8; ++r) {
        const int row = m0 + wm * 32 + sm * 16 + half * 8 + r;
        out[(size_t)row * N + col] = acc[sm][sn][r];
      }
    }
  }
}

// ---------------------------------------------------------------------------
extern "C" void kernel_launch(void* const* d_in, const int* in_sizes, int n_in,
                              void* d_out, int out_size, void* d_ws,
                              size_t ws_size, hipStream_t stream) {
  (void)in_sizes; (void)n_in; (void)out_size; (void)ws_size;

  const float* X  = (const float*)d_in[0];  // [4, 4096, 1024]
  const float* Wh = (const float*)d_in[1];  // [1024, 1536]
  const float* Wg = (const float*)d_in[2];  // [1024, 1536]
  const float* Wo = (const float*)d_in[3];  // [1536, 1024]
  float* out = (float*)d_out;               // [4, 4096, 1024]

  const int Bq = 4, S = 4096, D = 1024, Di = 1536;
  const int M = Bq * S;  // 16384

  float* lcb = (float*)d_ws;                  // log_coeffs, then h (in place)
  float* lvb = lcb + (size_t)M * Di;          // log_values

  dim3 blk(256);
  mingru_gemm_hg<<<dim3((M / TM) * (Di / TN)), blk, 0, stream>>>(
      X, Wh, Wg, lcb, lvb, M, D, Di);
  mingru_scan<<<dim3((Bq * Di) / 256), blk, 0, stream>>>(lcb, lvb, Bq, S, Di);
  mingru_gemm_out<<<dim3((M / TM) * (D / TN)), blk, 0, stream>>>(
      lcb, Wo, out, M, Di, D);
}